// Model_51024211476619
// MI455X (gfx1250) — compile-verified
//
#include <hip/hip_runtime.h>

#define WAVE 32

typedef __attribute__((ext_vector_type(2))) float v2f;
typedef __attribute__((ext_vector_type(8))) float v8f;

constexpr int Bb = 4;
constexpr int Vv = 5023;
constexpr int NBn = 10;
constexpr int KK = 9;
constexpr int M = Bb * Vv;            // 20092 rows
constexpr int VTILES = (M + 15) / 16; // 1256

// ---------------------------------------------------------------------------
// Pack W [Cin,K,Cout] and u [Cin,K] into Wcat [Cin x Npad] = [ W | u | 0pad ]
// ---------------------------------------------------------------------------
__global__ void pack_weights(const float* __restrict__ W, const float* __restrict__ u,
                             float* __restrict__ Wcat, int Cin, int Cout, int Npad) {
    int idx = blockIdx.x * blockDim.x + threadIdx.x;
    if (idx >= Cin * Npad) return;
    int ci = idx / Npad;
    int n  = idx % Npad;
    float val = 0.0f;
    if (n < KK * Cout) {
        int k = n / Cout, o = n % Cout;
        val = W[(ci * KK + k) * Cout + o];
    } else if (n < KK * Cout + KK) {
        val = u[ci * KK + (n - KK * Cout)];
    }
    Wcat[idx] = val;
}

// ---------------------------------------------------------------------------
// h0 = relu(x @ Win)   x:[M,3] Win:[3,16] -> h0:[M,16]
// ---------------------------------------------------------------------------
__global__ void input_layer(const float* __restrict__ x, const float* __restrict__ Win,
                            float* __restrict__ h) {
    int idx = blockIdx.x * blockDim.x + threadIdx.x;
    if (idx >= M * 16) return;
    int r = idx >> 4, o = idx & 15;
    float acc = 0.0f;
#pragma unroll
    for (int c = 0; c < 3; ++c) acc += x[r * 3 + c] * Win[c * 16 + o];
    h[idx] = fmaxf(acc, 0.0f);
}

// ---------------------------------------------------------------------------
// FP32 WMMA GEMM: out[M x Npad] = h[M x Cin] @ Wcat[Cin x Npad]
// One wave per (16-row, 16-col) tile; contraction over Cin in steps of 4 with
// V_WMMA_F32_16X16X4_F32.
//
// A frag (16x4 f32, 2 VGPR): lane l holds A[M = l&15][K = k0 + 2*(l>>4) + j]
// B frag (4x16 f32, 2 VGPR): lane l holds B[K = k0 + 2*(l>>4) + j][N = l&15]
// C/D   (16x16 f32, 8 VGPR): lane l, reg v holds D[M = v + 8*(l>>4)][N = l&15]
// ---------------------------------------------------------------------------
__global__ void gemm_wmma(const float* __restrict__ h, const float* __restrict__ Wcat,
                          float* __restrict__ out, int Cin, int Npad, int ntiles,
                          int total_waves) {
    int wid  = (int)((blockIdx.x * blockDim.x + threadIdx.x) / WAVE);
    int lane = threadIdx.x & (WAVE - 1);
    if (wid >= total_waves) return;   // wave-uniform exit (EXEC stays all-ones for WMMA)

    int vt = wid / ntiles;
    int nt = wid % ntiles;

    int mrow   = vt * 16 + (lane & 15);
    int mrow_c = mrow < M ? mrow : (M - 1);     // clamp loads; padded rows never stored
    int khalf  = (lane >> 4) * 2;
    int col    = nt * 16 + (lane & 15);

    v8f acc = {};
    for (int k0 = 0; k0 < Cin; k0 += 4) {
        v2f a, b;
        a.x = h[mrow_c * Cin + k0 + khalf + 0];
        a.y = h[mrow_c * Cin + k0 + khalf + 1];
        b.x = Wcat[(k0 + khalf + 0) * Npad + col];
        b.y = Wcat[(k0 + khalf + 1) * Npad + col];
        acc = __builtin_amdgcn_wmma_f32_16x16x4_f32(
            /*neg_a=*/false, a, /*neg_b=*/false, b,
            /*c_mod=*/(short)0, acc, /*reuse_a=*/false, /*reuse_b=*/false);
    }

#pragma unroll
    for (int v = 0; v < 8; ++v) {
        int r = vt * 16 + (lane >> 4) * 8 + v;
        if (r < M) out[(size_t)r * Npad + col] = acc[v];
    }
}

// ---------------------------------------------------------------------------
// Aggregation: per vertex, gather neighbor ux/wx, softmax over K, reduce,
// scale by 1/deg, add bias, ReLU. One wave per row; lane = output channel.
// wxu row layout: [ wx(k*Cout+o) for K*Cout | ux(k) for K | pad ]
// ---------------------------------------------------------------------------
__global__ void aggregate(const float* __restrict__ wxu, const int* __restrict__ adj,
                          const float* __restrict__ cvec, const float* __restrict__ bias,
                          float* __restrict__ hout, int Cout, int Npad) {
    int wid  = (int)((blockIdx.x * blockDim.x + threadIdx.x) / WAVE);
    int lane = threadIdx.x & (WAVE - 1);
    if (wid >= M) return;

    int b = wid / Vv;
    int v = wid % Vv;
    int oc = lane < Cout ? lane : (Cout - 1);   // clamp so padded lanes stay in-bounds

    const float* urow = wxu + (size_t)wid * Npad + KK * Cout;
    float us[KK], cb[KK];
#pragma unroll
    for (int k = 0; k < KK; ++k) { us[k] = urow[k]; cb[k] = cvec[k]; }

    float acc = 0.0f;
    int   cnt = 0;
    for (int n = 0; n < NBn; ++n) {
        int a = adj[v * NBn + n];           // wave-uniform
        if (a == 0) continue;               // zero-padded neighbor: contribution is 0
        ++cnt;
        const float* nrow = wxu + (size_t)(b * Vv + (a - 1)) * Npad;

        float t[KK];
        float mx = -1e30f;
#pragma unroll
        for (int k = 0; k < KK; ++k) {
            t[k] = us[k] + nrow[KK * Cout + k] + cb[k];
            mx = fmaxf(mx, t[k]);
        }
        float s = 0.0f;
#pragma unroll
        for (int k = 0; k < KK; ++k) { t[k] = __expf(t[k] - mx); s += t[k]; }
        float inv = 1.0f / s;
#pragma unroll
        for (int k = 0; k < KK; ++k) acc += (t[k] * inv) * nrow[k * Cout + oc];
    }

    float recip = cnt > 0 ? 1.0f / (float)cnt : 0.0f;
    float val   = acc * recip + bias[oc];
    if (lane < Cout) hout[(size_t)wid * Cout + lane] = fmaxf(val, 0.0f);
}

// ---------------------------------------------------------------------------
// out = h @ Wout   h:[M,3] Wout:[3,3]
// ---------------------------------------------------------------------------
__global__ void output_layer(const float* __restrict__ h, const float* __restrict__ Wout,
                             float* __restrict__ out) {
    int idx = blockIdx.x * blockDim.x + threadIdx.x;
    if (idx >= M * 3) return;
    int r = idx / 3, j = idx % 3;
    float acc = 0.0f;
#pragma unroll
    for (int c = 0; c < 3; ++c) acc += h[r * 3 + c] * Wout[c * 3 + j];
    out[idx] = acc;
}

// ---------------------------------------------------------------------------
extern "C" void kernel_launch(void* const* d_in, const int* in_sizes, int n_in,
                              void* d_out, int out_size, void* d_ws, size_t ws_size,
                              hipStream_t stream) {
    (void)in_sizes; (void)n_in; (void)out_size; (void)ws_size;

    const float* x    = (const float*)d_in[0];
    const int*   adj  = (const int*)d_in[1];
    const float* Win  = (const float*)d_in[2];
    const float* Wout = (const float*)d_in[3];

    // Layer configs
    const int Cin_[6]  = {16, 16, 16, 32, 16, 16};
    const int Cout_[6] = {16, 16, 32, 16, 16, 3};
    const int Npad_[6] = {160, 160, 304, 160, 160, 48}; // round16(K*Cout + K)

    // Workspace layout (floats)
    float* ws  = (float*)d_ws;
    float* hA  = ws;                       // M * 32
    float* hB  = hA + (size_t)M * 32;      // M * 32
    float* wxu = hB + (size_t)M * 32;      // M * 304 (max Npad)
    float* wcat[6];
    {
        float* p = wxu + (size_t)M * 304;
        for (int i = 0; i < 6; ++i) { wcat[i] = p; p += (size_t)Cin_[i] * Npad_[i]; }
    }

    // Pack per-layer weight matrices [W | u | pad]
    for (int i = 0; i < 6; ++i) {
        const float* W = (const float*)d_in[4 + 4 * i];
        const float* u = (const float*)d_in[6 + 4 * i];
        int total = Cin_[i] * Npad_[i];
        pack_weights<<<(total + 255) / 256, 256, 0, stream>>>(W, u, wcat[i],
                                                              Cin_[i], Cout_[i], Npad_[i]);
    }

    // Input projection
    input_layer<<<(M * 16 + 255) / 256, 256, 0, stream>>>(x, Win, hA);

    // 6 NLayers, ping-pong hA/hB
    float* hin  = hA;
    float* hout = hB;
    for (int i = 0; i < 6; ++i) {
        const float* bbias = (const float*)d_in[5 + 4 * i];
        const float* cvec  = (const float*)d_in[7 + 4 * i];

        int ntiles = Npad_[i] / 16;
        int total_waves = VTILES * ntiles;
        int gblocks = (total_waves * WAVE + 255) / 256;
        gemm_wmma<<<gblocks, 256, 0, stream>>>(hin, wcat[i], wxu,
                                               Cin_[i], Npad_[i], ntiles, total_waves);

        int ablocks = (M * WAVE + 255) / 256;
        aggregate<<<ablocks, 256, 0, stream>>>(wxu, adj, cvec, bbias, hout,
                                               Cout_[i], Npad_[i]);

        float* t = hin; hin = hout; hout = t;
    }

    // Final projection (last layer output is in hin after the swap)
    output_layer<<<(M * 3 + 255) / 256, 256, 0, stream>>>(hin, Wout, (float*)d_out);
}